// ProductLayer_83330955477234
// MI455X (gfx1250) — compile-verified
//
#include <hip/hip_runtime.h>

typedef __attribute__((ext_vector_type(2))) float v2f;
typedef __attribute__((ext_vector_type(4))) float v4f;
typedef __attribute__((ext_vector_type(8))) float v8f;
typedef int v4i_vs __attribute__((vector_size(16)));   // matches builtin param print

#define PL_AS1 __attribute__((address_space(1)))
#define PL_AS3 __attribute__((address_space(3)))

// ---- feature probes -------------------------------------------------------
#if defined(__has_builtin)
#if __has_builtin(__builtin_amdgcn_wmma_f32_16x16x4_f32)
#define PL_HAVE_WMMA 1
#endif
#if __has_builtin(__builtin_amdgcn_global_load_async_to_lds_b128)
#define PL_HAVE_ASYNC 1
#endif
#if __has_builtin(__builtin_amdgcn_s_wait_asynccnt)
#define PL_WAIT_ASYNC() __builtin_amdgcn_s_wait_asynccnt(0)
#else
#define PL_WAIT_ASYNC() asm volatile("s_wait_asynccnt 0" ::: "memory")
#endif
#endif

// ---- geometry -------------------------------------------------------------
constexpr int kD            = 256;                 // embed dim (row length)
constexpr int kRowsPerWave  = 16;
constexpr int kWavesPerBlk  = 2;
constexpr int kRowsPerBlk   = kRowsPerWave * kWavesPerBlk;     // 32
constexpr int kS            = 260;                 // padded LDS row stride (dwords): 260%64=4 -> conflict-free column reads
constexpr int kTileFloats   = kRowsPerWave * kS;   // 4160
constexpr int kWaveLds      = kTileFloats + 16;    // + 16 row sums

__global__ __launch_bounds__(kWavesPerBlk * 32)
void product_layer_rowscale(const float* __restrict__ x,
                            float* __restrict__ out,
                            int rows) {
    __shared__ float lds[kWavesPerBlk * kWaveLds];

    const int lane = threadIdx.x & 31;
    const int wave = threadIdx.x >> 5;

    float* wlds = lds + wave * kWaveLds;
    float* sums = wlds + kTileFloats;

    const int tile_row0 = (blockIdx.x * kWavesPerBlk + wave) * kRowsPerWave;
    if (tile_row0 >= rows) return;                       // wave-uniform

    const float* __restrict__ gsrc = x  + (size_t)tile_row0 * kD;
    float*       __restrict__ gdst = out + (size_t)tile_row0 * kD;

    // ---- ragged tail (wave-uniform branch; unused for 131072 rows) --------
    const int nrows = rows - tile_row0;
    if (nrows < kRowsPerWave) {
        for (int rr = 0; rr < nrows; ++rr) {
            const float* rp = gsrc + rr * kD;
            float s = 0.0f;
            for (int c = lane; c < kD; c += 32) s += rp[c];
            for (int off = 16; off; off >>= 1) s += __shfl_xor(s, off, 32);
            for (int c = lane; c < kD; c += 32) gdst[rr * kD + c] = rp[c] * s;
        }
        return;
    }

    // ---- stage 16x256 f32 tile -> LDS (padded rows), coalesced b128 -------
#if defined(PL_HAVE_ASYNC)
#pragma unroll
    for (int it = 0; it < 32; ++it) {
        const int flat = (it * 32 + lane) * 4;           // float index in tile
        const int row  = flat >> 8;
        const int col  = flat & 255;
        __builtin_amdgcn_global_load_async_to_lds_b128(
            (PL_AS1 v4i_vs*)(gsrc + flat),
            (PL_AS3 v4i_vs*)(wlds + row * kS + col),
            0, 0);
    }
    PL_WAIT_ASYNC();
    asm volatile("" ::: "memory");
#else
#pragma unroll
    for (int it = 0; it < 32; ++it) {
        const int flat = (it * 32 + lane) * 4;
        const int row  = flat >> 8;
        const int col  = flat & 255;
        *(v4f*)(wlds + row * kS + col) = *(const v4f*)(gsrc + flat);
    }
#endif

    // ---- row sums: rowsum = X(16x256) * ones(256x1), exact in f32 ---------
    const int  half = lane >> 4;          // 0: K=0,1   1: K=2,3 per A-chunk
    const int  r    = lane & 15;          // A-matrix row index M
    const float* abase = wlds + r * kS + half * 2;

#if defined(PL_HAVE_WMMA)
    v2f bones; bones.x = 1.0f; bones.y = 1.0f;
    v8f acc0 = {0.f,0.f,0.f,0.f,0.f,0.f,0.f,0.f};
    v8f acc1 = {0.f,0.f,0.f,0.f,0.f,0.f,0.f,0.f};
#pragma unroll
    for (int kc = 0; kc < 64; kc += 2) {
        v2f a0 = *(const v2f*)(abase + (kc + 0) * 4);    // ds_load_b64, conflict-free
        v2f a1 = *(const v2f*)(abase + (kc + 1) * 4);
        acc0 = __builtin_amdgcn_wmma_f32_16x16x4_f32(false, a0, false, bones,
                                                     (short)0, acc0, false, false);
        acc1 = __builtin_amdgcn_wmma_f32_16x16x4_f32(false, a1, false, bones,
                                                     (short)0, acc1, false, false);
    }
    v8f acc = acc0 + acc1;
    // D[m,n] == rowsum[m] for every n. Lane 0 holds rows 0..7 (VGPR v -> M=v),
    // lane 16 holds rows 8..15 (VGPR v -> M=v+8). Spill 16 sums to LDS.
    if (lane == 0 || lane == 16) {
        float* dstp = sums + (half ? 8 : 0);
#pragma unroll
        for (int v = 0; v < 8; ++v) dstp[v] = acc[v];
    }
#else
    if (lane < 16) {
        float s = 0.0f;
        for (int c = 0; c < kD; c += 4) {
            v4f t = *(const v4f*)(wlds + r * kS + c);
            s += t.x + t.y + t.z + t.w;
        }
        sums[r] = s;
    }
#endif

    // ---- scale + store, coalesced b128 ------------------------------------
#pragma unroll
    for (int it = 0; it < 32; ++it) {
        const int flat = (it * 32 + lane) * 4;
        const int row  = flat >> 8;
        const int col  = flat & 255;
        v4f t = *(const v4f*)(wlds + row * kS + col);
        const float sc = sums[row];                      // broadcast LDS read
        t *= sc;
        *(v4f*)(gdst + flat) = t;
    }
}

extern "C" void kernel_launch(void* const* d_in, const int* in_sizes, int n_in,
                              void* d_out, int out_size, void* d_ws, size_t ws_size,
                              hipStream_t stream) {
    (void)n_in; (void)out_size; (void)d_ws; (void)ws_size;
    const float* x   = (const float*)d_in[0];
    float*       out = (float*)d_out;
    const long long n    = (long long)in_sizes[0];
    const int       rows = (int)(n / kD);                // 131072 for [128,1024,256]
    const int       blocks = (rows + kRowsPerBlk - 1) / kRowsPerBlk;
    product_layer_rowscale<<<blocks, kWavesPerBlk * 32, 0, stream>>>(x, out, rows);
}